// EventWarping_11879879544374
// MI455X (gfx1250) — compile-verified
//
#include <hip/hip_runtime.h>
#include <hip/hip_bf16.h>
#include <stdint.h>

// Problem constants (from reference)
#define H_ 256
#define W_ 336
#define HW_ (H_ * W_)
#define B_ 8
#define N_ 262144            // events per batch (2^18)
#define FLOW_SCALE 336.0f    // max(H, W)
#define REG_WEIGHT 0.001f
#define EV_BLOCK 256

typedef __attribute__((ext_vector_type(4))) unsigned int u32x4;
typedef __attribute__((ext_vector_type(8))) int          i32x8;
typedef __attribute__((ext_vector_type(4))) int          i32x4;
typedef __attribute__((ext_vector_type(2))) float        v2f;
typedef __attribute__((ext_vector_type(8))) float        v8f;

// TDM path only on the device pass where the gfx1250 builtins exist.
#if defined(__HIP_DEVICE_COMPILE__) && defined(__gfx1250__) && \
    __has_builtin(__builtin_amdgcn_tensor_load_to_lds) && \
    __has_builtin(__builtin_amdgcn_s_wait_tensorcnt)
#define USE_TDM 1
#else
#define USE_TDM 0
#endif

#if defined(__HIP_DEVICE_COMPILE__) && defined(__gfx1250__) && \
    __has_builtin(__builtin_amdgcn_wmma_f32_16x16x4_f32)
#define USE_WMMA_REDUCE 1
#else
#define USE_WMMA_REDUCE 0
#endif

__device__ __forceinline__ void atomic_add_f32(float* p, float v) {
  // relaxed agent-scope fadd; result unused -> no-return global_atomic_add_f32
  __hip_atomic_fetch_add(p, v, __ATOMIC_RELAXED, __HIP_MEMORY_SCOPE_AGENT);
}

// Full-wave (32-lane) sum. WMMA variant: A(16x4) holds the lane values
// (lanes 0-15 in k=0, lanes 16-31 in k=2), B = ones(4x16), so
// D[m][n] = v_m + v_{m+16} replicated over n. Summing the 8 C/D VGPRs
// in-lane gives rows 0-7 (lanes 0-15) / rows 8-15 (lanes 16-31); one
// xor-shuffle completes the reduction, result in every lane.
__device__ __forceinline__ float wave_sum(float v) {
#if USE_WMMA_REDUCE
  v2f a; a.x = v;    a.y = 0.0f;   // A[m][0]=v_m, A[m][2]=v_{m+16}, odd k = 0
  v2f b; b.x = 1.0f; b.y = 1.0f;   // B = ones
  v8f c = {};
  c = __builtin_amdgcn_wmma_f32_16x16x4_f32(
      /*neg_a=*/false, a, /*neg_b=*/false, b,
      /*c_mod=*/(short)0, c, /*reuse_a=*/false, /*reuse_b=*/false);
  float t = ((c[0] + c[1]) + (c[2] + c[3])) + ((c[4] + c[5]) + (c[6] + c[7]));
  t += __shfl_xor(t, 16, 32);
  return t;
#else
#pragma unroll
  for (int o = 16; o > 0; o >>= 1) v += __shfl_down(v, o, 32);
  return v;
#endif
}

__device__ __forceinline__ float wave_sum_shfl(float v) {
#pragma unroll
  for (int o = 16; o > 0; o >>= 1) v += __shfl_down(v, o, 32);
  return v;
}

__device__ __forceinline__ void block_atomic_sum(float v, float* out, float scale) {
  __shared__ float wsum[8];
  const int lane = threadIdx.x & 31;
  const int wid  = threadIdx.x >> 5;
  v = wave_sum(v);  // all lanes active here -> EXEC all 1s for WMMA
  if (lane == 0) wsum[wid] = v;
  __syncthreads();
  if (wid == 0) {
    float s = (lane < (int)(blockDim.x >> 5)) ? wsum[lane] : 0.0f;
    s = wave_sum_shfl(s);
    if (lane == 0) atomic_add_f32(out, s * scale);
  }
}

// ---------------------------------------------------------------------------
// Kernel 0: zero the accumulator images (B128 stores) and the scalar output
// ---------------------------------------------------------------------------
__global__ __launch_bounds__(256) void zero_kernel(float4* __restrict__ acc,
                                                   float* __restrict__ out,
                                                   int total4) {
  const int stride = gridDim.x * blockDim.x;
  const float4 z = make_float4(0.f, 0.f, 0.f, 0.f);
  for (int i = blockIdx.x * blockDim.x + threadIdx.x; i < total4; i += stride)
    acc[i] = z;
  if (blockIdx.x == 0 && threadIdx.x == 0) out[0] = 0.0f;
}

// ---------------------------------------------------------------------------
// Kernel 1: per-event warp + bilinear scatter-add.
// acc layout: group g = ((b*2 + tref_idx)*2 + p), each group is HW float2
// pairs {iwe, iwe_ts} interleaved so both atomics per corner share a line.
// Event tile (256 * float4 = 4 KB) is staged into LDS by the Tensor Data
// Mover (one tensor_load_to_lds per workgroup, issued by wave 0).
// ---------------------------------------------------------------------------
__global__ __launch_bounds__(EV_BLOCK) void event_kernel(
    const float4* __restrict__ ev, const float* __restrict__ flow,
    float* __restrict__ acc) {
  const int gid = blockIdx.x * EV_BLOCK + threadIdx.x;

#if USE_TDM
  extern __shared__ float4 tile[];  // dynamic LDS -> lds_addr 0
  if (threadIdx.x < 32) {  // TDM is a per-wave DMA op: issue from wave 0 only
    const uint64_t ga = (uint64_t)(uintptr_t)(ev + (size_t)blockIdx.x * EV_BLOCK);
    // ---- D# group 0: count=1, lds_addr=0, 57-bit global addr, type=2 ----
    u32x4 g0;
    g0.x = 1u;                                   // count = 1 valid descriptor
    g0.y = 0u;                                   // lds_addr (bytes)
    g0.z = (unsigned)(ga & 0xFFFFFFFFu);         // global_addr[31:0]
    g0.w = (unsigned)((ga >> 32) & 0x01FFFFFFu)  // global_addr[56:32]
           | (2u << 30);                         // type = 2 ("image")
    // ---- D# group 1: data_size=4B, 1-D tile of EV_BLOCK*4 dwords ----
    const unsigned ndw = EV_BLOCK * 4u;  // 1024 dwords = 256 events * 16B
    const unsigned w0 = (2u << 16);                  // data_size = 4 bytes
    const unsigned w1 = (ndw & 0xFFFFu) << 16;       // tensor_dim0[15:0]
    const unsigned w2 = (ndw >> 16) & 0xFFFFu;       // tensor_dim0[31:16]
    const unsigned w3 = (ndw & 0xFFFFu) << 16;       // tile_dim0 = ndw
    const unsigned w4 = 0u;                          // tile_dim1/2 unused (1-D)
    const unsigned w5 = ndw;                         // tensor_dim0_stride
    i32x8 g1 = {(int)w0, (int)w1, (int)w2, (int)w3,
                (int)w4, (int)w5, 0, 0};
    i32x4 gz4 = {0, 0, 0, 0};          // groups 2/3 unused (<=2-D tensor)
    i32x8 gz8 = {0, 0, 0, 0, 0, 0, 0, 0};
    __builtin_amdgcn_tensor_load_to_lds(g0, g1, gz4, gz4, gz8, 0);
    __builtin_amdgcn_s_wait_tensorcnt(0);
  }
  __syncthreads();
  const float4 e = tile[threadIdx.x];
#else
  const float4 e = ev[gid];
#endif

  const int   b   = gid >> 18;  // N = 2^18 events per batch
  const float ts  = e.x;
  const float ys  = e.y;
  const float xs  = e.z;
  const int   p   = (e.w > 0.0f) ? 0 : 1;  // pol_mask = [pol==1, pol==0]

  const int fidx = (int)fmaf(ys, (float)W_, xs);  // ys,xs are integral
  const float* fb = flow + (size_t)b * 2 * HW_;
  const float flx = fb[fidx];        // channel 0 = x flow
  const float fly = fb[HW_ + fidx];  // channel 1 = y flow

#pragma unroll
  for (int t = 0; t < 2; ++t) {
    const float tref  = (t == 0) ? 1.0f : 0.0f;
    const float tgrad = (t == 0) ? ts : (1.0f - ts);
    const float dt = (tref - ts) * FLOW_SCALE;
    const float wy = fmaf(dt, fly, ys);
    const float wx = fmaf(dt, flx, xs);
    const float tyf = floorf(wy), lxf = floorf(wx);
    const float fy = wy - tyf, fx = wx - lxf;
    const int ty = (int)tyf, lx = (int)lxf;
    const float wys[2] = {1.0f - fy, fy};
    const float wxs[2] = {1.0f - fx, fx};
    float* base = acc + (size_t)(((b * 2 + t) * 2 + p)) * (size_t)(2 * HW_);
#pragma unroll
    for (int dy = 0; dy < 2; ++dy) {
      const int cy = ty + dy;
      if ((unsigned)cy >= (unsigned)H_) continue;
#pragma unroll
      for (int dx = 0; dx < 2; ++dx) {
        const int cx = lx + dx;
        if ((unsigned)cx >= (unsigned)W_) continue;
        const float w = wys[dy] * wxs[dx];
        float* pix = base + (size_t)(cy * W_ + cx) * 2;
        atomic_add_f32(pix, w);             // iwe
        atomic_add_f32(pix + 1, w * tgrad); // iwe_ts
      }
    }
  }
}

// ---------------------------------------------------------------------------
// Kernel 2: loss = sum((iwe_ts / (iwe + 1e-9))^2); B128 loads carry two
// {iwe, iwe_ts} pairs each.
// ---------------------------------------------------------------------------
__global__ __launch_bounds__(256) void loss_kernel(const float4* __restrict__ acc,
                                                   float* __restrict__ out,
                                                   int n4) {
  float s = 0.0f;
  const int stride = gridDim.x * blockDim.x;
  for (int i = blockIdx.x * blockDim.x + threadIdx.x; i < n4; i += stride) {
    const float4 v = acc[i];
    const float r0 = v.y / (v.x + 1e-9f);
    const float r1 = v.w / (v.z + 1e-9f);
    s = fmaf(r0, r0, s);
    s = fmaf(r1, r1, s);
  }
  block_atomic_sum(s, out, 1.0f);
}

// ---------------------------------------------------------------------------
// Kernel 3: Charbonnier smoothness over flow (B,2,H,W)
// ---------------------------------------------------------------------------
__global__ __launch_bounds__(256) void smooth_kernel(const float* __restrict__ flow,
                                                     float* __restrict__ out) {
  const int total = B_ * 2 * H_ * W_;
  float s = 0.0f;
  const int stride = gridDim.x * blockDim.x;
  for (int i = blockIdx.x * blockDim.x + threadIdx.x; i < total; i += stride) {
    const int w = i % W_;
    const int h = (i / W_) % H_;
    const float f = flow[i];
    if (h + 1 < H_) { const float d = f - flow[i + W_]; s += sqrtf(fmaf(d, d, 1e-6f)); }
    if (w + 1 < W_) { const float d = f - flow[i + 1];  s += sqrtf(fmaf(d, d, 1e-6f)); }
  }
  block_atomic_sum(s, out, REG_WEIGHT);
}

// ---------------------------------------------------------------------------
extern "C" void kernel_launch(void* const* d_in, const int* in_sizes, int n_in,
                              void* d_out, int out_size, void* d_ws, size_t ws_size,
                              hipStream_t stream) {
  (void)in_sizes; (void)n_in; (void)out_size; (void)ws_size;
  const float*  flow   = (const float*)d_in[0];
  const float4* events = (const float4*)d_in[1];
  // d_in[2] (pol_mask) is redundant: polarity is event_list[...,3] > 0.
  float* out = (float*)d_out;
  float* acc = (float*)d_ws;  // B*4 images * HW * {iwe, iwe_ts} = 21 MB

  const int accN = B_ * 4 * HW_ * 2;  // 5,505,024 floats (divisible by 4)
  zero_kernel<<<1024, 256, 0, stream>>>((float4*)acc, out, accN / 4);

  const int evBlocks = (B_ * N_) / EV_BLOCK;  // 8192
  event_kernel<<<evBlocks, EV_BLOCK, EV_BLOCK * sizeof(float4), stream>>>(
      events, flow, acc);

  loss_kernel<<<2048, 256, 0, stream>>>((const float4*)acc, out,
                                        (B_ * 4 * HW_) / 2);
  smooth_kernel<<<1024, 256, 0, stream>>>(flow, out);
}